// BotanHadamardTransform_8959301779558
// MI455X (gfx1250) — compile-verified
//
#include <hip/hip_runtime.h>

typedef float v2f __attribute__((ext_vector_type(2)));
typedef float v8f __attribute__((ext_vector_type(8)));

#define HN 4096
#define ROWS_PER_BLOCK 4   // 4 waves/block, one row per wave, 64 KB static LDS

// y[row] = FWHT_4096(x[row]) / 64, computed as three Kronecker stages of
// 16x16 H16 matmuls using v_wmma_f32_16x16x4_f32 (4 chained K=4 WMMAs per
// 16x16x16 tile). H16 entries are (+/-)0.25 generated from parity(i&j).
// All global traffic is fully coalesced b128; the WMMA-layout scatter/gather
// lives entirely in LDS.
__global__ __launch_bounds__(128) void fwht4096_wmma_kernel(
    const float* __restrict__ x, float* __restrict__ y) {
  __shared__ float lds[ROWS_PER_BLOCK * HN];  // 64 KB

  const int tid  = threadIdx.x;
  const int wave = tid >> 5;
  const int lane = tid & 31;
  const int half = lane >> 4;   // which 16-lane half
  const int lo   = lane & 15;

  float* buf = lds + wave * HN;
  const int row = blockIdx.x * ROWS_PER_BLOCK + wave;
  const float* __restrict__ xin  = x + (size_t)row * HN;
  float*       __restrict__ yout = y + (size_t)row * HN;

  // ---- coalesced fill: row -> LDS (32 x float4 per lane) ----
  {
    const float4* __restrict__ src = (const float4*)xin;
    float4* dst = (float4*)buf;
#pragma unroll
    for (int j = 0; j < 32; ++j)
      dst[j * 32 + lane] = src[j * 32 + lane];
  }

  // ---- H16 operand registers (scaled by 0.25 per stage; H is symmetric so
  //      the same registers serve as both A-layout and B-layout operands) ----
  v2f h[4];
#pragma unroll
  for (int i = 0; i < 4; ++i) {
    const int k = 4 * i + 2 * half;
    h[i].x = (__builtin_popcount(lo & k) & 1) ? -0.25f : 0.25f;
    h[i].y = (__builtin_popcount(lo & (k + 1)) & 1) ? -0.25f : 0.25f;
  }

  __syncthreads();

  // ---- Stage 1: contract c.  D[b][c'] = sum_c T[a][b][c] * H[c][c'] ----
  for (int a = 0; a < 16; ++a) {
    float* t = buf + a * 256;
    v2f A[4];
#pragma unroll
    for (int i = 0; i < 4; ++i) {
      const int k = 4 * i + 2 * half;
      A[i].x = t[lo * 16 + k];
      A[i].y = t[lo * 16 + k + 1];
    }
    v8f c = {0.f, 0.f, 0.f, 0.f, 0.f, 0.f, 0.f, 0.f};
#pragma unroll
    for (int i = 0; i < 4; ++i)
      c = __builtin_amdgcn_wmma_f32_16x16x4_f32(false, A[i], false, h[i],
                                                (short)0, c, false, false);
#pragma unroll
    for (int v = 0; v < 8; ++v)
      t[(v + 8 * half) * 16 + lo] = c[v];
  }

  __syncthreads();

  // ---- Stage 2: contract b.  D[b'][c'] = sum_b H[b'][b] * T1[a][b][c'] ----
  for (int a = 0; a < 16; ++a) {
    float* t = buf + a * 256;
    v2f B[4];
#pragma unroll
    for (int i = 0; i < 4; ++i) {
      const int k = 4 * i + 2 * half;
      B[i].x = t[k * 16 + lo];
      B[i].y = t[(k + 1) * 16 + lo];
    }
    v8f c = {0.f, 0.f, 0.f, 0.f, 0.f, 0.f, 0.f, 0.f};
#pragma unroll
    for (int i = 0; i < 4; ++i)
      c = __builtin_amdgcn_wmma_f32_16x16x4_f32(false, h[i], false, B[i],
                                                (short)0, c, false, false);
#pragma unroll
    for (int v = 0; v < 8; ++v)
      t[(v + 8 * half) * 16 + lo] = c[v];
  }

  __syncthreads();

  // ---- Stage 3: contract a; write D back to LDS in-place ----
  for (int g = 0; g < 16; ++g) {
    v2f B[4];
#pragma unroll
    for (int i = 0; i < 4; ++i) {
      const int k = 4 * i + 2 * half;
      B[i].x = buf[k * 256 + g * 16 + lo];
      B[i].y = buf[(k + 1) * 256 + g * 16 + lo];
    }
    v8f c = {0.f, 0.f, 0.f, 0.f, 0.f, 0.f, 0.f, 0.f};
#pragma unroll
    for (int i = 0; i < 4; ++i)
      c = __builtin_amdgcn_wmma_f32_16x16x4_f32(false, h[i], false, B[i],
                                                (short)0, c, false, false);
#pragma unroll
    for (int v = 0; v < 8; ++v)
      buf[(v + 8 * half) * 256 + g * 16 + lo] = c[v];
  }

  __syncthreads();

  // ---- coalesced drain: LDS -> row (32 x float4 per lane, full 128B lines) ----
  {
    const float4* src = (const float4*)buf;
    float4* __restrict__ dst = (float4*)yout;
#pragma unroll
    for (int j = 0; j < 32; ++j)
      dst[j * 32 + lane] = src[j * 32 + lane];
  }
}

extern "C" void kernel_launch(void* const* d_in, const int* in_sizes, int n_in,
                              void* d_out, int out_size, void* d_ws, size_t ws_size,
                              hipStream_t stream) {
  const float* x = (const float*)d_in[0];
  // d_in[1] (H) is not read: Hadamard signs are synthesized as parity(i&j).
  float* y = (float*)d_out;

  const int nrows  = in_sizes[0] / HN;            // 4*4096 = 16384
  const int blocks = nrows / ROWS_PER_BLOCK;      // 4096
  hipLaunchKernelGGL(fwht4096_wmma_kernel, dim3(blocks), dim3(128), 0, stream,
                     x, y);
}